// GridPull_14233521619389
// MI455X (gfx1250) — compile-verified
//
#include <hip/hip_runtime.h>
#include <hip/hip_bf16.h>

// grid_pull: trilinear interpolation, dct2 (reflect) boundary, extrapolate.
//   x:    (1, 2, 192, 192, 192) float32
//   grid: (1, 192, 192, 192, 3) float32, voxel coords in [-2, n+1]
//   out:  (1, 2, 192, 192, 192) float32
//
// MI455X strategy (bandwidth-bound gather, ~200 MB HBM floor ≈ 8.6 us):
//  - x (56.6 MB) fits in the 192 MB L2; random gathers become L2 hits.
//  - grid/out are pure streams -> non-temporal hints so they don't evict x.
//  - global_prefetch_b8 runs the grid stream ahead.
//  - one thread per voxel computes BOTH channels (reuse indices + weights).

namespace {
constexpr int Wv = 192;
constexpr int Hv = 192;
constexpr int Dv = 192;
constexpr int NVOX = Wv * Hv * Dv;          // 7,077,888 voxels
constexpr int BLOCK = 256;                   // 8 wave32 per workgroup
constexpr int NBLK = (NVOX + BLOCK - 1) / BLOCK;
}

// dct2 / reflect boundary: symmetric about -0.5 and n-0.5, period 2n.
// Matches jnp: i = mod(i, 2n) (nonnegative); i >= n ? 2n-1-i : i.
__device__ __forceinline__ int reflect_dct2(int i, int n) {
    const int p = 2 * n;                 // 384: constant -> mulhi sequence
    i %= p;
    i += (i < 0) ? p : 0;                // nonnegative remainder
    return (i < n) ? i : (p - 1 - i);
}

__global__ __launch_bounds__(BLOCK)
void grid_pull_dct2_kernel(const float* __restrict__ x,
                           const float* __restrict__ grid,
                           float* __restrict__ out) {
    const int n = blockIdx.x * BLOCK + threadIdx.x;
    if (n >= NVOX) return;

    // ---- prefetch the grid stream ~192 KB ahead (3 cachelines per wave,
    //      issued by lanes 0..2; no dep-counter cost on gfx1250) ----
    {
        const int lane = threadIdx.x & 31;
        if (lane < 3) {
            // this wave consumes grid bytes [waveBase, waveBase + 384)
            const long waveBase = (long)(n & ~31) * 12;
            const long pf = waveBase + 192 * 1024 + (long)lane * 128;
            if (pf < (long)NVOX * 12)
                __builtin_prefetch((const char*)grid + pf, 0, 1);
        }
    }

    // ---- load this voxel's target coordinate (streaming, non-temporal) ----
    const size_t g3 = (size_t)n * 3;
    const float gx = __builtin_nontemporal_load(grid + g3 + 0);
    const float gy = __builtin_nontemporal_load(grid + g3 + 1);
    const float gz = __builtin_nontemporal_load(grid + g3 + 2);

    const float flx = floorf(gx), fly = floorf(gy), flz = floorf(gz);
    const int lox = (int)flx, loy = (int)fly, loz = (int)flz;
    const float fx1 = gx - flx, fy1 = gy - fly, fz1 = gz - flz;
    const float fx0 = 1.0f - fx1, fy0 = 1.0f - fy1, fz0 = 1.0f - fz1;

    // reflected neighbor indices
    const int ix0 = reflect_dct2(lox,     Wv);
    const int ix1 = reflect_dct2(lox + 1, Wv);
    const int iy0 = reflect_dct2(loy,     Hv);
    const int iy1 = reflect_dct2(loy + 1, Hv);
    const int iz0 = reflect_dct2(loz,     Dv);
    const int iz1 = reflect_dct2(loz + 1, Dv);

    // row bases for the 4 (x,y) combinations
    const int r00 = (ix0 * Hv + iy0) * Dv;
    const int r01 = (ix0 * Hv + iy1) * Dv;
    const int r10 = (ix1 * Hv + iy0) * Dv;
    const int r11 = (ix1 * Hv + iy1) * Dv;

    // (x,y) plane weights
    const float w00 = fx0 * fy0;
    const float w01 = fx0 * fy1;
    const float w10 = fx1 * fy0;
    const float w11 = fx1 * fy1;

    // ---- gather 8 taps for both channels (regular-temporal: keep x in L2) --
    const float* __restrict__ xc0 = x;
    const float* __restrict__ xc1 = x + NVOX;
    float a0 = 0.0f, a1 = 0.0f;

#define GP_TAP(rowBase, izv, wgt)                            \
    do {                                                     \
        const int _i = (rowBase) + (izv);                    \
        const float _w = (wgt);                              \
        a0 = fmaf(xc0[_i], _w, a0);                          \
        a1 = fmaf(xc1[_i], _w, a1);                          \
    } while (0)

    GP_TAP(r00, iz0, w00 * fz0);
    GP_TAP(r00, iz1, w00 * fz1);
    GP_TAP(r01, iz0, w01 * fz0);
    GP_TAP(r01, iz1, w01 * fz1);
    GP_TAP(r10, iz0, w10 * fz0);
    GP_TAP(r10, iz1, w10 * fz1);
    GP_TAP(r11, iz0, w11 * fz0);
    GP_TAP(r11, iz1, w11 * fz1);
#undef GP_TAP

    // ---- streaming stores (non-temporal: don't evict x from L2) ----
    __builtin_nontemporal_store(a0, out + n);
    __builtin_nontemporal_store(a1, out + NVOX + n);
}

extern "C" void kernel_launch(void* const* d_in, const int* in_sizes, int n_in,
                              void* d_out, int out_size, void* d_ws, size_t ws_size,
                              hipStream_t stream) {
    (void)in_sizes; (void)n_in; (void)out_size; (void)d_ws; (void)ws_size;
    const float* x    = (const float*)d_in[0];   // (1,2,192,192,192)
    const float* grid = (const float*)d_in[1];   // (1,192,192,192,3)
    float* out        = (float*)d_out;           // (1,2,192,192,192)

    grid_pull_dct2_kernel<<<NBLK, BLOCK, 0, stream>>>(x, grid, out);
}